// BatchedKDE_8143257993222
// MI455X (gfx1250) — compile-verified
//
#include <hip/hip_runtime.h>
#include <math.h>

typedef float v2f __attribute__((ext_vector_type(2)));
typedef float v8f __attribute__((ext_vector_type(8)));

#define NB   4
#define NQ   4096
#define NM   4096
#define ND   32
#define PERB (NQ * ND)          // 131072 elems per batch
#define TOT  (NB * PERB)        // 524288 total X_fit elems
#define L2E  1.4426950408889634f

// workspace layout (bytes)
#define HIST2_OFF_U32 65536                  // hist1: [0,65536) u32
#define SUMS_BYTE     (65536 * 5 * 4)        // after hist1(256KB)+hist2(1MB) = 1310720
#define SEL_BYTE      (SUMS_BYTE + 64)       // 8 x u32 : selBin[4], selRank[4]
#define INVBW_BYTE    (SEL_BYTE + 32)        // 4 x f32
#define ZERO_BYTES    (SUMS_BYTE + 64)       // zero hists + sums

// ---- order-preserving float<->uint mapping -------------------------------
__device__ __forceinline__ unsigned mapf(float f) {
    unsigned x = __float_as_uint(f);
    return (x & 0x80000000u) ? ~x : (x | 0x80000000u);
}
__device__ __forceinline__ float unmapf(unsigned u) {
    unsigned x = (u & 0x80000000u) ? (u ^ 0x80000000u) : ~u;
    return __uint_as_float(x);
}

// ---- pass 1: 16-bit-MSB histogram + per-batch sum / sumsq ----------------
__global__ __launch_bounds__(256) void kde_stats(const float* __restrict__ xf,
                                                 unsigned* __restrict__ hist1,
                                                 double* __restrict__ sums) {
    __shared__ double sd[256], sq[256];
    const int t = threadIdx.x;
    const long long base = ((long long)blockIdx.x * 256 + t) * 4;
    const float4 v = *(const float4*)(xf + base);
    const float arr[4] = {v.x, v.y, v.z, v.w};
    double s = 0.0, q = 0.0;
#pragma unroll
    for (int i = 0; i < 4; ++i) {
        atomicAdd(&hist1[mapf(arr[i]) >> 16], 1u);
        const double d = (double)arr[i];
        s += d; q += d * d;
    }
    sd[t] = s; sq[t] = q;
    __syncthreads();
    for (int off = 128; off > 0; off >>= 1) {
        if (t < off) { sd[t] += sd[t + off]; sq[t] += sq[t + off]; }
        __syncthreads();
    }
    if (t == 0) {
        const int batch = blockIdx.x >> 7;   // 128 blocks per batch (1024 elems/block)
        atomicAdd(&sums[2 * batch],     sd[0]);
        atomicAdd(&sums[2 * batch + 1], sq[0]);
    }
}

// ---- parallel k-th selection over a 65536-bin histogram (1024 threads) ---
__device__ void select_kth(const unsigned* __restrict__ hist, unsigned k,
                           unsigned* outBin, unsigned* outRank, unsigned* sh) {
    const int t = threadIdx.x;               // 1024 threads, 64 bins each
    const unsigned* hp = hist + t * 64;
    unsigned part = 0;
    for (int i = 0; i < 64; ++i) part += hp[i];
    __syncthreads();                          // protect sh reuse across calls
    sh[t] = part;
    __syncthreads();
    for (int off = 1; off < 1024; off <<= 1) {  // Hillis-Steele inclusive scan
        unsigned v = (t >= off) ? sh[t - off] : 0u;
        __syncthreads();
        sh[t] += v;
        __syncthreads();
    }
    const unsigned excl = sh[t] - part;
    if (k >= excl && k < excl + part) {       // owner thread walks its bins
        unsigned c = excl;
        for (int i = 0; i < 64; ++i) {
            const unsigned n = hp[i];
            if (k < c + n) { *outBin = (unsigned)(t * 64 + i); *outRank = k - c; break; }
            c += n;
        }
    }
}

__global__ __launch_bounds__(1024) void kde_scan1(const unsigned* __restrict__ hist1,
                                                  unsigned* __restrict__ sel) {
    __shared__ unsigned sh[1024];
    // (n-1)*0.25 = 131071.75 ; (n-1)*0.75 = 393215.25
    const unsigned ks[4] = {131071u, 131072u, 393215u, 393216u};
    for (int j = 0; j < 4; ++j)
        select_kth(hist1, ks[j], &sel[j], &sel[4 + j], sh);
}

// ---- pass 2: low-16-bit histograms restricted to selected MSB bins -------
__global__ __launch_bounds__(256) void kde_hist2(const float* __restrict__ xf,
                                                 const unsigned* __restrict__ sel,
                                                 unsigned* __restrict__ hist2) {
    const unsigned b0 = sel[0], b1 = sel[1], b2 = sel[2], b3 = sel[3];
    const long long base = ((long long)blockIdx.x * 256 + threadIdx.x) * 4;
    const float4 v = *(const float4*)(xf + base);
    const float arr[4] = {v.x, v.y, v.z, v.w};
#pragma unroll
    for (int i = 0; i < 4; ++i) {
        const unsigned u = mapf(arr[i]);
        const unsigned top = u >> 16, low = u & 0xFFFFu;
        if (top == b0) atomicAdd(&hist2[0 * 65536 + low], 1u);
        if (top == b1) atomicAdd(&hist2[1 * 65536 + low], 1u);
        if (top == b2) atomicAdd(&hist2[2 * 65536 + low], 1u);
        if (top == b3) atomicAdd(&hist2[3 * 65536 + low], 1u);
    }
}

// ---- finalize: order statistics -> q25/q75 -> per-batch 1/bw -------------
__global__ __launch_bounds__(1024) void kde_final(const unsigned* __restrict__ hist2,
                                                  const unsigned* __restrict__ sel,
                                                  const double* __restrict__ sums,
                                                  float* __restrict__ invbw) {
    __shared__ unsigned sh[1024];
    __shared__ unsigned shLow[4], shR[4];
    for (int j = 0; j < 4; ++j)
        select_kth(hist2 + j * 65536, sel[4 + j], &shLow[j], &shR[j], sh);
    __syncthreads();
    if (threadIdx.x == 0) {
        float ost[4];
        for (int j = 0; j < 4; ++j)
            ost[j] = unmapf((sel[j] << 16) | shLow[j]);
        const float q25 = ost[0] + 0.75f * (ost[1] - ost[0]);
        const float q75 = ost[2] + 0.25f * (ost[3] - ost[2]);
        const float iqr134 = (q75 - q25) / 1.34f;
        const double N = (double)PERB;
        for (int b = 0; b < NB; ++b) {
            const double s = sums[2 * b], q = sums[2 * b + 1];
            const double var = (q - s * s / N) / (N - 1.0);
            const float sd = (float)sqrt(var);
            // 0.9 / 4096^0.2 ; 4096^0.2 = 2^2.4
            const float bw = (float)(0.9 / 5.2780316430915766) * fminf(sd, iqr134);
            invbw[b] = 1.0f / bw;
        }
    }
}

// ---- main fused kernel: f32 WMMA GEMM + exp2-sum epilogue ----------------
// grid = 128 blocks (4 batches x 32 query tiles of 128 rows), 256 thr = 8 waves.
// Each wave owns a 16-row query tile held in registers across the whole M loop.
// exp(-dist/bw) = exp2((2*dot - nx2 - nmu2) * invbw * log2e); log2e is folded
// into the precomputed coefficients so the epilogue per output row is exactly
// one v_add + one v_fma + one v_exp_f32 + one v_add.
#define MC 256   // X_fit rows staged in LDS per chunk (32 KB)
__global__ __launch_bounds__(256) void kde_main(const float* __restrict__ xq,
                                                const float* __restrict__ xf,
                                                const float* __restrict__ invbw_p,
                                                float* __restrict__ out) {
    __shared__ float sB[MC * ND];     // 32 KB fit-row chunk
    __shared__ float sMu[MC];         // nmu2[row] * invbw * log2e for the chunk
    __shared__ float sNx[128];        // per-block nx2 broadcast
    const int tid   = threadIdx.x;
    const int w     = tid >> 5, lane = tid & 31;
    const int batch = blockIdx.x >> 5;
    const int n0    = (blockIdx.x & 31) * 128;
    const int r     = lane & 15;          // row within 16-tile == N column in C
    const int kb    = (lane >> 4) << 1;   // k offset: 0 for lanes 0-15, 2 for 16-31
    const int hi    = lane >> 4;
    const float ibwl  = invbw_p[batch] * L2E;   // invbw * log2e
    const float ibwl2 = 2.0f * ibwl;

    // A tile: query rows n0+w*16 .. +15, all K=32 -> 8 x v2f per lane
    const float* qrow = xq + ((size_t)batch * NQ + n0 + w * 16 + r) * ND;
    v2f a[8];
    float s2 = 0.f;
#pragma unroll
    for (int s = 0; s < 8; ++s) {
        a[s] = *(const v2f*)(qrow + 4 * s + kb);
        s2 += a[s].x * a[s].x + a[s].y * a[s].y;
    }
    s2 += __shfl_xor(s2, 16, 32);         // lanes r and r+16 jointly hold all K
    if (lane < 16) sNx[w * 16 + r] = s2;  // nx2 of this wave's 16 rows
    __syncthreads();
    float nxs[8];                          // nx2 * invbw * log2e, per output row j
#pragma unroll
    for (int j = 0; j < 8; ++j) nxs[j] = sNx[w * 16 + j + 8 * hi] * ibwl;

    float acc[8] = {0.f, 0.f, 0.f, 0.f, 0.f, 0.f, 0.f, 0.f};
    const float* fbase = xf + (size_t)batch * NM * ND;

    for (int ch = 0; ch < NM / MC; ++ch) {
        __syncthreads();
        // staging: thread t owns fit-row t of the chunk (8 x float4 = 32 floats);
        // squares it in registers -> sMu[t] = nmu2 * invbw * log2e, then stores.
        const float4* src = (const float4*)(fbase + (size_t)ch * MC * ND);
        float4 rbuf[8];
#pragma unroll
        for (int i = 0; i < 8; ++i) rbuf[i] = src[tid * 8 + i];
        float ss = 0.f;
#pragma unroll
        for (int i = 0; i < 8; ++i)
            ss += rbuf[i].x * rbuf[i].x + rbuf[i].y * rbuf[i].y +
                  rbuf[i].z * rbuf[i].z + rbuf[i].w * rbuf[i].w;
#pragma unroll
        for (int i = 0; i < 8; ++i) ((float4*)sB)[tid * 8 + i] = rbuf[i];
        sMu[tid] = ss * ibwl;
        if (ch + 1 < NM / MC)             // global_prefetch_b8 of next chunk
            __builtin_prefetch(fbase + (size_t)(ch + 1) * MC * ND + tid * 32, 0, 0);
        __syncthreads();

        for (int mt = 0; mt < MC / 16; ++mt) {
            const float* frow = &sB[(mt * 16 + r) * ND];
            const float t2s = sMu[mt * 16 + r];   // nmu2[m]*invbw*l2e at N=lane&15
            v8f c = {};
#pragma unroll
            for (int s = 0; s < 8; ++s) {   // K=32 as 8 x WMMA f32 16x16x4
                const v2f bv = *(const v2f*)(frow + 4 * s + kb);
                c = __builtin_amdgcn_wmma_f32_16x16x4_f32(
                        false, a[s], false, bv, (short)0, c, false, false);
            }
#pragma unroll
            for (int j = 0; j < 8; ++j)     // exp2 of pre-scaled argument
                acc[j] += __builtin_amdgcn_exp2f(c[j] * ibwl2 - (nxs[j] + t2s));
        }
    }

    // reduce over fit columns = lanes within each 16-lane half
#pragma unroll
    for (int j = 0; j < 8; ++j) {
        float v = acc[j];
        v += __shfl_xor(v, 1, 32);
        v += __shfl_xor(v, 2, 32);
        v += __shfl_xor(v, 4, 32);
        v += __shfl_xor(v, 8, 32);
        acc[j] = v;
    }
    float* orow = out + (size_t)batch * NQ + n0 + w * 16;
    if (lane == 0) {                       // rows j (C layout: lanes 0-15 -> M=j)
#pragma unroll
        for (int j = 0; j < 8; ++j) orow[j] = acc[j];
    } else if (lane == 16) {               // rows j+8 (lanes 16-31 -> M=j+8)
#pragma unroll
        for (int j = 0; j < 8; ++j) orow[8 + j] = acc[j];
    }
}

extern "C" void kernel_launch(void* const* d_in, const int* in_sizes, int n_in,
                              void* d_out, int out_size, void* d_ws, size_t ws_size,
                              hipStream_t stream) {
    const float* xq = (const float*)d_in[0];   // X_query [4,4096,32] f32
    const float* xf = (const float*)d_in[1];   // X_fit   [4,4096,32] f32
    float* out = (float*)d_out;                // [4,4096] f32

    unsigned* ws32  = (unsigned*)d_ws;
    unsigned* hist1 = ws32;                                   // 65536 u32
    unsigned* hist2 = ws32 + HIST2_OFF_U32;                   // 4 x 65536 u32
    double*   sums  = (double*)((char*)d_ws + SUMS_BYTE);     // 4 x {sum,sumsq}
    unsigned* sel   = (unsigned*)((char*)d_ws + SEL_BYTE);    // bin[4], rank[4]
    float*    invbw = (float*)((char*)d_ws + INVBW_BYTE);     // 4 x 1/bw

    hipMemsetAsync(d_ws, 0, ZERO_BYTES, stream);              // zero hists + sums

    kde_stats<<<TOT / 1024, 256, 0, stream>>>(xf, hist1, sums);
    kde_scan1<<<1, 1024, 0, stream>>>(hist1, sel);
    kde_hist2<<<TOT / 1024, 256, 0, stream>>>(xf, sel, hist2);
    kde_final<<<1, 1024, 0, stream>>>(hist2, sel, sums, invbw);
    kde_main<<<NB * (NQ / 128), 256, 0, stream>>>(xq, xf, invbw, out);
}